// MultiheadVolAttn3d_27109833573058
// MI455X (gfx1250) — compile-verified
//
#include <hip/hip_runtime.h>

typedef float v2f __attribute__((ext_vector_type(2)));
typedef float v8f __attribute__((ext_vector_type(8)));

#define VS 262144u      // 64*64*64, voxel stride between channels (elements)
#define LROW 20         // padded LDS row stride (floats) to avoid bank conflicts

__device__ __forceinline__ v8f wmma_f32(v2f a, v2f b, v8f c) {
  // D = A(16x4) x B(4x16) + C, fp32, wave32
  return __builtin_amdgcn_wmma_f32_16x16x4_f32(false, a, false, b, (short)0, c, false, false);
}

// Precompute Weff = Wout @ blockdiag(Whead), beff = bout + Wout @ bhead_flat
__global__ void build_weff_kernel(const float* __restrict__ whead, const float* __restrict__ bhead,
                                  const float* __restrict__ wout,  const float* __restrict__ bout,
                                  float* __restrict__ weff, float* __restrict__ beff) {
  const int o = blockIdx.x;        // 0..63 output channel
  const int c = threadIdx.x;       // 0..63 input channel (of ctx)
  const int h = c >> 4, cc = c & 15;
  float s = 0.f;
  for (int j = 0; j < 16; ++j)
    s += wout[o*64 + h*16 + j] * whead[h*256 + j*16 + cc];
  weff[o*64 + c] = s;
  if (c == 0) {
    float b = bout[o];
    for (int m = 0; m < 64; ++m) b += wout[o*64 + m] * bhead[m];
    beff[o] = b;
  }
}

__global__ __launch_bounds__(256) void volattn_kernel(
    const float* __restrict__ Xq, const float* __restrict__ Xk, const float* __restrict__ Xv,
    const float* __restrict__ wq, const float* __restrict__ bq,
    const float* __restrict__ wk, const float* __restrict__ bk,
    const float* __restrict__ wv, const float* __restrict__ bv,
    const float* __restrict__ weff, const float* __restrict__ beff,
    float* __restrict__ out)
{
  const unsigned lane = threadIdx.x & 31;
  const unsigned wid  = threadIdx.x >> 5;
  const unsigned nn   = lane & 15;   // column (voxel within tile) for B/C/D fragments
  const unsigned hi   = lane >> 4;   // lane-half select
  const unsigned col  = (blockIdx.x * 8u + wid) * 16u + nn;   // global voxel index (fits 32-bit)

  __shared__ float lds[8][64*LROW];
  float* my = lds[wid];

  v2f bf[16];  // staged B fragments for one 64x16 activation tile (K=64 -> 16 steps of 4)

  // Streaming (read-once) loads: uniform SGPR base + 32-bit per-lane offset, non-temporal
  auto stageB = [&](const float* __restrict__ X) {
#pragma unroll
    for (unsigned kt = 0; kt < 16; ++kt) {
      const unsigned k0 = 4u*kt + 2u*hi;          // B layout: VGPR0 rows {k,k+2}, VGPR1 rows {k+1,k+3}
      bf[kt].x = __builtin_nontemporal_load(X + (k0      )*VS + col);
      bf[kt].y = __builtin_nontemporal_load(X + (k0 + 1u )*VS + col);
    }
  };

  // acc[mt] = (W @ Xtile + bias) rows [16*mt, 16*mt+16), columns = this wave's 16 voxels
  auto proj = [&](const float* __restrict__ W, const float* __restrict__ bias, v8f* acc) {
#pragma unroll
    for (unsigned mt = 0; mt < 4; ++mt) {
      v8f c = {0.f,0.f,0.f,0.f,0.f,0.f,0.f,0.f};
#pragma unroll
      for (unsigned kt = 0; kt < 16; ++kt) {
        // A layout: lane row = lane&15, K pair at 2*(lane>>4); contiguous float2 in row-major W
        v2f a = *(const v2f*)(W + (mt*16u + nn)*64u + 4u*kt + 2u*hi);
        c = wmma_f32(a, bf[kt], c);
      }
#pragma unroll
      for (unsigned r = 0; r < 8; ++r) c[r] += bias[mt*16u + 8u*hi + r];
      acc[mt] = c;
    }
  };

  // ---- Q projection (kept live) ----
  v8f qacc[4];
  stageB(Xq);
  proj(wq, bq, qacc);

  // ---- K projections + per-voxel per-head scores ----
  float sc[4][4];                       // [l][h]
  for (int l = 0; l < 4; ++l) {
    v8f kacc[4];
    stageB(Xk + (unsigned long)l*64ul*VS);
    proj(wk, bk, kacc);
#pragma unroll
    for (int h = 0; h < 4; ++h) {       // head h == m-tile h (channels 16h..16h+15)
      float p = 0.f;
#pragma unroll
      for (int r = 0; r < 8; ++r) p += qacc[h][r]*kacc[h][r];
      p += __shfl_xor(p, 16, 32);       // combine the two lane-halves (channels +8)
      sc[l][h] = p * 0.25f;             // 1/sqrt(16)
    }
  }

  // ---- softmax over L=4 context elements ----
  float attn[4][4];
#pragma unroll
  for (int h = 0; h < 4; ++h) {
    float m = fmaxf(fmaxf(sc[0][h], sc[1][h]), fmaxf(sc[2][h], sc[3][h]));
    float s = 0.f;
#pragma unroll
    for (int l = 0; l < 4; ++l) { attn[l][h] = __expf(sc[l][h] - m); s += attn[l][h]; }
    const float inv = 1.0f / s;
#pragma unroll
    for (int l = 0; l < 4; ++l) attn[l][h] *= inv;
  }

  // ---- V projections, weighted accumulation into ctx ----
  v8f ctx[4] = {};
  for (int l = 0; l < 4; ++l) {
    v8f vacc[4];
    stageB(Xv + (unsigned long)l*64ul*VS);
    proj(wv, bv, vacc);
#pragma unroll
    for (int h = 0; h < 4; ++h)
#pragma unroll
      for (int r = 0; r < 8; ++r) ctx[h][r] = fmaf(attn[l][h], vacc[h][r], ctx[h][r]);
  }

  // ---- transpose ctx (C/D layout -> B layout) through LDS ----
#pragma unroll
  for (unsigned h = 0; h < 4; ++h)
#pragma unroll
    for (unsigned r = 0; r < 8; ++r)
      my[(16u*h + 8u*hi + r)*LROW + nn] = ctx[h][r];

  __syncthreads();

#pragma unroll
  for (unsigned kt = 0; kt < 16; ++kt) {
    const unsigned k0 = 4u*kt + 2u*hi;
    bf[kt].x = my[(k0     )*LROW + nn];
    bf[kt].y = my[(k0 + 1u)*LROW + nn];
  }

  // ---- fused (Wout @ blockdiag(Whead)) GEMM + bias, non-temporal store ----
#pragma unroll
  for (unsigned mt = 0; mt < 4; ++mt) {
    v8f c = {0.f,0.f,0.f,0.f,0.f,0.f,0.f,0.f};
#pragma unroll
    for (unsigned kt = 0; kt < 16; ++kt) {
      v2f a = *(const v2f*)(weff + (mt*16u + nn)*64u + 4u*kt + 2u*hi);
      c = wmma_f32(a, bf[kt], c);
    }
#pragma unroll
    for (unsigned r = 0; r < 8; ++r)
      __builtin_nontemporal_store(c[r] + beff[mt*16u + 8u*hi + r],
                                  out + (mt*16u + 8u*hi + r)*VS + col);
  }
}

extern "C" void kernel_launch(void* const* d_in, const int* in_sizes, int n_in,
                              void* d_out, int out_size, void* d_ws, size_t ws_size,
                              hipStream_t stream) {
  (void)in_sizes; (void)n_in; (void)out_size; (void)ws_size;
  const float* query  = (const float*)d_in[0];
  const float* keys   = (const float*)d_in[1];
  const float* values = (const float*)d_in[2];
  const float* wq   = (const float*)d_in[3];
  const float* bq   = (const float*)d_in[4];
  const float* wk   = (const float*)d_in[5];
  const float* bk   = (const float*)d_in[6];
  const float* wv   = (const float*)d_in[7];
  const float* bv   = (const float*)d_in[8];
  const float* whead = (const float*)d_in[9];
  const float* bhead = (const float*)d_in[10];
  const float* wout  = (const float*)d_in[11];
  const float* bout  = (const float*)d_in[12];

  float* weff = (float*)d_ws;          // 64*64 floats
  float* beff = weff + 64*64;          // 64 floats

  build_weff_kernel<<<64, 64, 0, stream>>>(whead, bhead, wout, bout, weff, beff);

  // 262144 voxels / (8 waves * 16 voxels) = 2048 blocks
  volattn_kernel<<<2048, 256, 0, stream>>>(query, keys, values,
                                           wq, bq, wk, bk, wv, bv,
                                           weff, beff, (float*)d_out);
}